// Grouped_query_17179869184428
// MI455X (gfx1250) — compile-verified
//
#include <hip/hip_runtime.h>
#include <hip/hip_bf16.h>
#include <math.h>

#define N 6144
#define TILE 16
#define NTILES (N / TILE)   // 384
#define WPB 8               // waves per block (wave32)
#define BLOCK (WPB * 32)    // 256 threads
#define LOG2E 1.44269504088896340736f

typedef __attribute__((ext_vector_type(2))) float v2f;
typedef __attribute__((ext_vector_type(8))) float v8f;

// single-instruction maxes (avoid cmp+cndmask and NaN-canonicalization lowering)
__device__ __forceinline__ float vmaxf(float a, float b) {
  float d;
  asm("v_max_num_f32 %0, %1, %2" : "=v"(d) : "v"(a), "v"(b));
  return d;
}
__device__ __forceinline__ float vmax3f(float a, float b, float c) {
  float d;
  asm("v_max3_num_f32 %0, %1, %2, %3" : "=v"(d) : "v"(a), "v"(b), "v"(c));
  return d;
}

// ---------------------------------------------------------------------------
// y = W x + b for 3x3 row-major W (torch Linear: y = x @ W^T + b)
// ---------------------------------------------------------------------------
__device__ __forceinline__ void lin3(const float* __restrict__ W,
                                     const float* __restrict__ B,
                                     float x0, float x1, float x2,
                                     float* o) {
  o[0] = fmaf(W[0], x0, fmaf(W[1], x1, fmaf(W[2], x2, B[0])));
  o[1] = fmaf(W[3], x0, fmaf(W[4], x1, fmaf(W[5], x2, B[1])));
  o[2] = fmaf(W[6], x0, fmaf(W[7], x1, fmaf(W[8], x2, B[2])));
}

__device__ __forceinline__ v2f pack2(float a, float b) {
  v2f f; f[0] = a; f[1] = b; return f;
}

// ---------------------------------------------------------------------------
// Kernel 1: all eight Linear(3,3) projections.
// G and K are stored directly in V_WMMA_F32_16X16X4_F32 fragment layout:
//   frag[2j+0] = (c0, c1)   -> lanes 0-15  (K=0 in VGPR0, K=1 in VGPR1)
//   frag[2j+1] = (c2, 0 )   -> lanes 16-31 (K=2 in VGPR0, K=3 pad)
// so the hot loops do a single strided b64 load per fragment, no selects.
// G is pre-scaled by log2(e): e^(g.k) == exp2((g*log2e).k) -> raw v_exp_f32.
// ---------------------------------------------------------------------------
__global__ void __launch_bounds__(256)
prep_kernel(const float* __restrict__ x,
            const float* __restrict__ wq1, const float* __restrict__ bq1,
            const float* __restrict__ wq2, const float* __restrict__ bq2,
            const float* __restrict__ wq3, const float* __restrict__ bq3,
            const float* __restrict__ wq4, const float* __restrict__ bq4,
            const float* __restrict__ wk1, const float* __restrict__ bk1,
            const float* __restrict__ wk2, const float* __restrict__ bk2,
            const float* __restrict__ wv1, const float* __restrict__ bv1,
            const float* __restrict__ wv2, const float* __restrict__ bv2,
            v2f* __restrict__ G1f, v2f* __restrict__ G2f,
            v2f* __restrict__ K1f, v2f* __restrict__ K2f,
            float4* __restrict__ V1, float4* __restrict__ V2) {
  int i = blockIdx.x * blockDim.x + threadIdx.x;
  if (i >= N) return;
  float x0 = x[3 * i + 0], x1 = x[3 * i + 1], x2 = x[3 * i + 2];
  float q1[3], q2[3], q3[3], q4[3], k1[3], k2[3], v1[3], v2[3];
  lin3(wq1, bq1, x0, x1, x2, q1);
  lin3(wq2, bq2, x0, x1, x2, q2);
  lin3(wq3, bq3, x0, x1, x2, q3);
  lin3(wq4, bq4, x0, x1, x2, q4);
  lin3(wk1, bk1, x0, x1, x2, k1);
  lin3(wk2, bk2, x0, x1, x2, k2);
  lin3(wv1, bv1, x0, x1, x2, v1);
  lin3(wv2, bv2, x0, x1, x2, v2);
  G1f[2 * i + 0] = pack2((q1[0] + q2[0]) * LOG2E, (q1[1] + q2[1]) * LOG2E);
  G1f[2 * i + 1] = pack2((q1[2] + q2[2]) * LOG2E, 0.f);
  G2f[2 * i + 0] = pack2((q3[0] + q4[0]) * LOG2E, (q3[1] + q4[1]) * LOG2E);
  G2f[2 * i + 1] = pack2((q3[2] + q4[2]) * LOG2E, 0.f);
  K1f[2 * i + 0] = pack2(k1[0], k1[1]);
  K1f[2 * i + 1] = pack2(k1[2], 0.f);
  K2f[2 * i + 0] = pack2(k2[0], k2[1]);
  K2f[2 * i + 1] = pack2(k2[2], 0.f);
  V1[i] = make_float4(v1[0], v1[1], v1[2], 0.f);
  V2[i] = make_float4(v2[0], v2[1], v2[2], 0.f);
}

// ---------------------------------------------------------------------------
// Kernel 2: one block per 16-row tile.  WMMA 16x16x4 score tiles; 3 passes:
//   A: row max (2x unrolled, v_max3_num_f32)
//   B: C-operand = -m  ->  e = v_exp(S) directly; row sums + Sum(e*v) both
//      heads; emit b.  a2 never touches memory.
//   C: C-operand = -(m + log2(sum))  ->  a1 element = one v_exp + one store.
// ---------------------------------------------------------------------------
__global__ void __launch_bounds__(BLOCK)
attn_kernel(const v2f* __restrict__ G1f, const v2f* __restrict__ G2f,
            const v2f* __restrict__ K1f, const v2f* __restrict__ K2f,
            const float4* __restrict__ V1, const float4* __restrict__ V2,
            const float* __restrict__ Wb, const float* __restrict__ bb,
            float* __restrict__ a1_out, float* __restrict__ b_out) {
  __shared__ float s_max1[WPB][16], s_max2[WPB][16];
  __shared__ float s_sum1[WPB][16], s_sum2[WPB][16];
  __shared__ float s_acc1[WPB][16][3], s_acc2[WPB][16][3];
  __shared__ float f_m1[16], f_m2[16], f_l1[16];

  const int tid  = threadIdx.x;
  const int wave = tid >> 5;
  const int lane = tid & 31;
  const int hi   = lane >> 4;   // 0: rows 0-7 / K 0-1, 1: rows 8-15 / K 2-3
  const int lcol = lane & 15;
  const int r0   = blockIdx.x * TILE;
  const int fb   = 2 * lcol + hi;   // per-lane fragment index base

  // A fragments for this block's 16 rows (loaded once).
  v2f A1 = G1f[2 * (r0 + lcol) + hi];
  v2f A2 = G2f[2 * (r0 + lcol) + hi];

  // ------------------------- Pass A: row maxima ---------------------------
  float rm1[8], rm2[8];
#pragma unroll
  for (int r = 0; r < 8; ++r) { rm1[r] = -3.402823466e38f; rm2[r] = -3.402823466e38f; }

  for (int t = wave; t < NTILES; t += 2 * WPB) {   // NTILES/WPB = 48 (even)
    int i0 = fb + 32 * t;
    int i1 = fb + 32 * (t + WPB);
    v2f B1a = K1f[i0], B2a = K2f[i0];
    v2f B1b = K1f[i1], B2b = K2f[i1];
    v8f z = {};
    v8f S1a = __builtin_amdgcn_wmma_f32_16x16x4_f32(false, A1, false, B1a, (short)0, z, false, false);
    v8f S2a = __builtin_amdgcn_wmma_f32_16x16x4_f32(false, A2, false, B2a, (short)0, z, false, false);
    v8f S1b = __builtin_amdgcn_wmma_f32_16x16x4_f32(false, A1, false, B1b, (short)0, z, false, false);
    v8f S2b = __builtin_amdgcn_wmma_f32_16x16x4_f32(false, A2, false, B2b, (short)0, z, false, false);
#pragma unroll
    for (int r = 0; r < 8; ++r) {
      rm1[r] = vmax3f(rm1[r], S1a[r], S1b[r]);
      rm2[r] = vmax3f(rm2[r], S2a[r], S2b[r]);
    }
  }
  // reduce across the 16 lanes of each half (offsets < 16 keep halves separate)
#pragma unroll
  for (int off = 1; off < 16; off <<= 1) {
#pragma unroll
    for (int r = 0; r < 8; ++r) {
      rm1[r] = vmaxf(rm1[r], __shfl_xor(rm1[r], off, 32));
      rm2[r] = vmaxf(rm2[r], __shfl_xor(rm2[r], off, 32));
    }
  }
  if (lcol == 0) {
#pragma unroll
    for (int r = 0; r < 8; ++r) {
      s_max1[wave][hi * 8 + r] = rm1[r];
      s_max2[wave][hi * 8 + r] = rm2[r];
    }
  }
  __syncthreads();
  if (tid < 16) {
    float m1 = s_max1[0][tid], m2 = s_max2[0][tid];
    for (int w = 1; w < WPB; ++w) {
      m1 = vmaxf(m1, s_max1[w][tid]);
      m2 = vmaxf(m2, s_max2[w][tid]);
    }
    f_m1[tid] = m1;
    f_m2[tid] = m2;
  }
  __syncthreads();

  v8f Cm1, Cm2;
#pragma unroll
  for (int r = 0; r < 8; ++r) {
    Cm1[r] = -f_m1[hi * 8 + r];
    Cm2[r] = -f_m2[hi * 8 + r];
  }

  // --------------- Pass B: row sums + Sum(e*v), emit b --------------------
  float ss1[8], ss2[8], ac1[8][3], ac2[8][3];
#pragma unroll
  for (int r = 0; r < 8; ++r) {
    ss1[r] = 0.f; ss2[r] = 0.f;
#pragma unroll
    for (int c = 0; c < 3; ++c) { ac1[r][c] = 0.f; ac2[r][c] = 0.f; }
  }

  for (int t = wave; t < NTILES; t += WPB) {
    int fi = fb + 32 * t;
    int c  = t * TILE + lcol;
    v2f B1 = K1f[fi];
    v2f B2 = K2f[fi];
    // C operand carries -rowmax: matrix unit emits (s' - m') directly.
    v8f S1 = __builtin_amdgcn_wmma_f32_16x16x4_f32(false, A1, false, B1, (short)0, Cm1, false, false);
    v8f S2 = __builtin_amdgcn_wmma_f32_16x16x4_f32(false, A2, false, B2, (short)0, Cm2, false, false);
    float4 w1 = V1[c];
    float4 w2 = V2[c];
#pragma unroll
    for (int r = 0; r < 8; ++r) {
      float e1 = __builtin_amdgcn_exp2f(S1[r]);
      float e2 = __builtin_amdgcn_exp2f(S2[r]);
      ss1[r] += e1;  ss2[r] += e2;
      ac1[r][0] = fmaf(e1, w1.x, ac1[r][0]);
      ac1[r][1] = fmaf(e1, w1.y, ac1[r][1]);
      ac1[r][2] = fmaf(e1, w1.z, ac1[r][2]);
      ac2[r][0] = fmaf(e2, w2.x, ac2[r][0]);
      ac2[r][1] = fmaf(e2, w2.y, ac2[r][1]);
      ac2[r][2] = fmaf(e2, w2.z, ac2[r][2]);
    }
  }
#pragma unroll
  for (int off = 1; off < 16; off <<= 1) {
#pragma unroll
    for (int r = 0; r < 8; ++r) {
      ss1[r] += __shfl_xor(ss1[r], off, 32);
      ss2[r] += __shfl_xor(ss2[r], off, 32);
#pragma unroll
      for (int c = 0; c < 3; ++c) {
        ac1[r][c] += __shfl_xor(ac1[r][c], off, 32);
        ac2[r][c] += __shfl_xor(ac2[r][c], off, 32);
      }
    }
  }
  if (lcol == 0) {
#pragma unroll
    for (int r = 0; r < 8; ++r) {
      int row = hi * 8 + r;
      s_sum1[wave][row] = ss1[r];
      s_sum2[wave][row] = ss2[r];
#pragma unroll
      for (int c = 0; c < 3; ++c) {
        s_acc1[wave][row][c] = ac1[r][c];
        s_acc2[wave][row][c] = ac2[r][c];
      }
    }
  }
  __syncthreads();
  if (tid < 16) {
    float t1 = 0.f, t2 = 0.f, a1c[3] = {0.f, 0.f, 0.f}, a2c[3] = {0.f, 0.f, 0.f};
    for (int w = 0; w < WPB; ++w) {
      t1 += s_sum1[w][tid];
      t2 += s_sum2[w][tid];
      for (int c = 0; c < 3; ++c) { a1c[c] += s_acc1[w][tid][c]; a2c[c] += s_acc2[w][tid][c]; }
    }
    float i1 = 1.f / t1, i2 = 1.f / t2;
    f_l1[tid] = __builtin_amdgcn_logf(t1);   // log2(sum1), folded into pass-C exponent
    float p0 = fmaf(a1c[0], i1, a2c[0] * i2);
    float p1 = fmaf(a1c[1], i1, a2c[1] * i2);
    float p2 = fmaf(a1c[2], i1, a2c[2] * i2);
    size_t ob = (size_t)(r0 + tid) * 3;
    b_out[ob + 0] = fmaf(Wb[0], p0, fmaf(Wb[1], p1, fmaf(Wb[2], p2, bb[0])));
    b_out[ob + 1] = fmaf(Wb[3], p0, fmaf(Wb[4], p1, fmaf(Wb[5], p2, bb[1])));
    b_out[ob + 2] = fmaf(Wb[6], p0, fmaf(Wb[7], p1, fmaf(Wb[8], p2, bb[2])));
  }
  __syncthreads();

  // ----------------------- Pass C: stream out a1 --------------------------
  // a1[i][j] = exp2(s' - m' - log2(sum)); the whole shift lives in the
  // WMMA C operand, so each output element costs one v_exp_f32 + one store.
  v8f Cc1;
#pragma unroll
  for (int r = 0; r < 8; ++r) Cc1[r] = Cm1[r] - f_l1[hi * 8 + r];

  const size_t rowbase = (size_t)(r0 + hi * 8) * N + (size_t)lcol;
  for (int t = wave; t < NTILES; t += WPB) {
    v2f B1 = K1f[fb + 32 * t];
    v8f S1 = __builtin_amdgcn_wmma_f32_16x16x4_f32(false, A1, false, B1, (short)0, Cc1, false, false);
    float* p = a1_out + rowbase + (size_t)t * TILE;
#pragma unroll
    for (int r = 0; r < 8; ++r) {
      p[(size_t)r * N] = __builtin_amdgcn_exp2f(S1[r]);
    }
  }
}

// ---------------------------------------------------------------------------
extern "C" void kernel_launch(void* const* d_in, const int* in_sizes, int n_in,
                              void* d_out, int out_size, void* d_ws, size_t ws_size,
                              hipStream_t stream) {
  const float* x   = (const float*)d_in[0];
  const float* wq1 = (const float*)d_in[1];  const float* bq1 = (const float*)d_in[2];
  const float* wq2 = (const float*)d_in[3];  const float* bq2 = (const float*)d_in[4];
  const float* wq3 = (const float*)d_in[5];  const float* bq3 = (const float*)d_in[6];
  const float* wq4 = (const float*)d_in[7];  const float* bq4 = (const float*)d_in[8];
  const float* wk1 = (const float*)d_in[9];  const float* bk1 = (const float*)d_in[10];
  const float* wk2 = (const float*)d_in[11]; const float* bk2 = (const float*)d_in[12];
  const float* wv1 = (const float*)d_in[13]; const float* bv1 = (const float*)d_in[14];
  const float* wv2 = (const float*)d_in[15]; const float* bv2 = (const float*)d_in[16];
  const float* wb  = (const float*)d_in[17]; const float* bbv = (const float*)d_in[18];

  // workspace layout: 4 fragment arrays (2N v2f each) + 2 float4 arrays = 96B/row
  v2f* G1f = (v2f*)d_ws;
  v2f* G2f = G1f + 2 * (size_t)N;
  v2f* K1f = G2f + 2 * (size_t)N;
  v2f* K2f = K1f + 2 * (size_t)N;
  float4* V1 = (float4*)(K2f + 2 * (size_t)N);
  float4* V2 = V1 + (size_t)N;

  float* out   = (float*)d_out;
  float* a1    = out;                          // [N, N]
  float* b_out = out + (size_t)N * (size_t)N;  // [N, 3]

  prep_kernel<<<(N + 255) / 256, 256, 0, stream>>>(
      x, wq1, bq1, wq2, bq2, wq3, bq3, wq4, bq4,
      wk1, bk1, wk2, bk2, wv1, bv1, wv2, bv2,
      G1f, G2f, K1f, K2f, V1, V2);

  attn_kernel<<<NTILES, BLOCK, 0, stream>>>(
      G1f, G2f, K1f, K2f, V1, V2, wb, bbv, a1, b_out);
}